// TSDFSampler_88364657148062
// MI455X (gfx1250) — compile-verified
//
#include <hip/hip_runtime.h>

// TSDF sampler: morton3d(clamped grid coords) gather from a 512^3 float grid.
// Memory-bound irregular gather; optimized for MI455X (gfx1250, wave32):
//  - 4 points/thread -> 3x global_load_b128 + 1x global_store_b128, coalesced
//  - non-temporal TH hints on the touch-once streams (xyzs in, tsdfs out) so
//    the 192MB L2 is reserved for the 512MB grid gather working set
//  - gather loads keep default RT policy (cross-wave cacheline reuse)

typedef float v4f __attribute__((ext_vector_type(4)));

#define GRID_M1 511.0f

__device__ __forceinline__ unsigned expand_bits(unsigned v) {
    v = (v * 0x00010001u) & 0xFF0000FFu;
    v = (v * 0x00000101u) & 0x0F00F00Fu;
    v = (v * 0x00000011u) & 0xC30C30C3u;
    v = (v * 0x00000005u) & 0x49249249u;
    return v;
}

__device__ __forceinline__ unsigned to_coord(float x) {
    // Matches: clip(0.5*(x+1)*(g-1), 0, g-1).astype(uint32)  (truncation)
    float c = 0.5f * (x + 1.0f);
    c = c * GRID_M1;
    c = fminf(fmaxf(c, 0.0f), GRID_M1);
    return (unsigned)c;  // v_cvt_u32_f32: truncate toward zero, x >= 0
}

__device__ __forceinline__ unsigned morton3d(float fx, float fy, float fz) {
    unsigned xx = expand_bits(to_coord(fx));
    unsigned yy = expand_bits(to_coord(fy));
    unsigned zz = expand_bits(to_coord(fz));
    return (xx << 2) | (yy << 1) | zz;
}

__global__ __launch_bounds__(256) void tsdf_sample4_kernel(
    const float* __restrict__ xyzs, const float* __restrict__ grid,
    float* __restrict__ out, int ngroups) {
    int t = blockIdx.x * blockDim.x + threadIdx.x;
    if (t >= ngroups) return;

    // 4 points = 12 floats = 48 bytes, 16B-aligned (48*t % 16 == 0).
    const float* p = xyzs + (size_t)t * 12u;
    v4f a = __builtin_nontemporal_load((const v4f*)(p + 0));  // x0 y0 z0 x1
    v4f b = __builtin_nontemporal_load((const v4f*)(p + 4));  // y1 z1 x2 y2
    v4f c = __builtin_nontemporal_load((const v4f*)(p + 8));  // z2 x3 y3 z3

    unsigned i0 = morton3d(a[0], a[1], a[2]);
    unsigned i1 = morton3d(a[3], b[0], b[1]);
    unsigned i2 = morton3d(b[2], b[3], c[0]);
    unsigned i3 = morton3d(c[1], c[2], c[3]);

    // Irregular gather: default (RT) temporal policy — grid lines are shared
    // across threads, let them live in L2.
    v4f r;
    r[0] = grid[i0];
    r[1] = grid[i1];
    r[2] = grid[i2];
    r[3] = grid[i3];

    __builtin_nontemporal_store(r, (v4f*)(out + (size_t)t * 4u));
}

// Tail handler for n % 4 != 0 (not hit for N_POINTS = 4194304, but keeps the
// launch correct for any n).
__global__ void tsdf_sample_tail_kernel(
    const float* __restrict__ xyzs, const float* __restrict__ grid,
    float* __restrict__ out, int start, int n) {
    int i = start + blockIdx.x * blockDim.x + threadIdx.x;
    if (i >= n) return;
    const float* p = xyzs + (size_t)i * 3u;
    unsigned m = morton3d(p[0], p[1], p[2]);
    out[i] = grid[m];
}

extern "C" void kernel_launch(void* const* d_in, const int* in_sizes, int n_in,
                              void* d_out, int out_size, void* d_ws, size_t ws_size,
                              hipStream_t stream) {
    (void)n_in; (void)d_ws; (void)ws_size; (void)out_size;
    const float* xyzs = (const float*)d_in[0];   // [N, 3] float32
    const float* grid = (const float*)d_in[1];   // [512^3] float32
    float* out = (float*)d_out;                  // [N] float32

    int n = in_sizes[0] / 3;       // number of points
    int ngroups = n / 4;           // 4 points per thread
    int rem_start = ngroups * 4;

    if (ngroups > 0) {
        int blocks = (ngroups + 255) / 256;
        tsdf_sample4_kernel<<<blocks, 256, 0, stream>>>(xyzs, grid, out, ngroups);
    }
    if (rem_start < n) {
        int rem = n - rem_start;
        int blocks = (rem + 255) / 256;
        tsdf_sample_tail_kernel<<<blocks, 256, 0, stream>>>(xyzs, grid, out,
                                                            rem_start, n);
    }
}